// SparseConv_B_90555090469375
// MI455X (gfx1250) — compile-verified
//
#include <hip/hip_runtime.h>

// CDNA5 / gfx1250, wave32. V_WMMA_F32_16X16X4_F32: A=16x4 f32 (2 VGPR),
// B=4x16 f32 (2 VGPR), C/D=16x16 f32 (8 VGPR).
typedef float v2f __attribute__((ext_vector_type(2)));
typedef float v8f __attribute__((ext_vector_type(8)));

// out[r, i, j] = x[r,i]*S1[i,j] + x[r,j]*S2[i,j]
//   S2 = mask (pooling cells), S1 = mask + I
// Expressed as GEMM: out[R x 4096] = x[R x 64] @ G[64 x 4096], done with
// f32 WMMA so all products are exact (x * {0,1}).
//
// Block = 256 threads (8 waves), handles 16 output rows r0..r0+15.
// Wave w handles i-rows [8w, 8w+8), all four 16-wide j tiles each.

__device__ __forceinline__ float mask_ij(int i, int j) {
    const int d = j - i;
    const bool m = (d >= 1 && d <= 15)
                || (d >= 17 && d <= 31 && (d & 1) == 1 && (i & 1) == 0)
                || (d >= 35 && d <= 63 && (d & 3) == 3 && (i & 3) == 0);
    return m ? 1.0f : 0.0f;
}

__global__ __launch_bounds__(256) void sparseconv_b_wmma(
        const float* __restrict__ x, float* __restrict__ out) {
    const int lane  = threadIdx.x & 31;
    const int wave  = threadIdx.x >> 5;
    const int n     = lane & 15;   // column within tile / A-row selector
    const int khalf = lane >> 4;   // 0: K pair {0,1}; 1: K pair {2,3}
    const int r0    = blockIdx.x << 4;   // 16 output rows per block

    // ---- Preload all 16 A fragments: aF[kb] = x[r0 + n, 4kb + 2*khalf + {0,1}]
    // Matches ISA 16x4 f32 A layout (VGPR0: K0|K2 across lane halves, VGPR1: K1|K3).
    const float* pbase = x + (size_t)(r0 + n) * 64 + (khalf << 1);
    v2f aF[16];
#pragma unroll
    for (int kb = 0; kb < 16; ++kb)
        aF[kb] = *(const v2f*)(pbase + (kb << 2));

    for (int t = 0; t < 8; ++t) {                  // i-row index (wave-uniform)
        const int i = (wave << 3) + t;             // 0..63
        // A fragment for the K-block containing i (dynamic kb -> reload, L1/L2 hit)
        const v2f ai = *(const v2f*)(pbase + ((i >> 2) << 2));
        const int ksel = (i & 3);                  // which K of that block equals i

#pragma unroll
        for (int q = 0; q < 4; ++q) {              // j tile: columns [16q, 16q+16)
            const int j  = (q << 4) + n;
            const float mf = mask_ij(i, j);
            const float dg = (i == j) ? 1.0f : 0.0f;

            v8f acc = {};
            // Four K-blocks covering the j range (kb compile-time after unroll)
#pragma unroll
            for (int qq = 0; qq < 4; ++qq) {
                const int kb = (q << 2) + qq;
                const int k0 = (kb << 2) + (khalf << 1);
                v2f b;
                b.x = ((k0     == i) ? (mf + dg) : 0.0f) + ((k0     == j) ? mf : 0.0f);
                b.y = ((k0 + 1 == i) ? (mf + dg) : 0.0f) + ((k0 + 1 == j) ? mf : 0.0f);
                acc = __builtin_amdgcn_wmma_f32_16x16x4_f32(
                        false, aF[kb], false, b, (short)0, acc, false, false);
            }
            // K-block containing i, if not already covered (wave-uniform branch,
            // EXEC stays all-ones). Here k==j is impossible and i!=j.
            if ((i >> 4) != q) {
                v2f b;
                b.x = (ksel == (khalf << 1))     ? mf : 0.0f;
                b.y = (ksel == (khalf << 1) + 1) ? mf : 0.0f;
                acc = __builtin_amdgcn_wmma_f32_16x16x4_f32(
                        false, ai, false, b, (short)0, acc, false, false);
            }

            // C/D layout: VGPR v -> M=v (lanes 0-15), M=v+8 (lanes 16-31), N=n.
            // Streaming (non-temporal) stores: output is write-once, 256 MB.
            float* po = out + ((size_t)(r0 + (khalf << 3)) << 12) + (i << 6) + (q << 4) + n;
#pragma unroll
            for (int v = 0; v < 8; ++v)
                __builtin_nontemporal_store(acc[v], po + ((size_t)v << 12));
        }
    }
}

extern "C" void kernel_launch(void* const* d_in, const int* in_sizes, int n_in,
                              void* d_out, int out_size, void* d_ws, size_t ws_size,
                              hipStream_t stream) {
    (void)n_in; (void)out_size; (void)d_ws; (void)ws_size;
    const float* x = (const float*)d_in[0];
    float* out = (float*)d_out;
    const int rows = in_sizes[0] / 64;        // B*D = 16384
    dim3 grid(rows / 16), block(256);
    hipLaunchKernelGGL(sparseconv_b_wmma, grid, block, 0, stream, x, out);
}